// DGG_LearnableK_SDD_10617159156342
// MI455X (gfx1250) — compile-verified
//
#include <hip/hip_runtime.h>

typedef __attribute__((ext_vector_type(2))) float v2f;
typedef __attribute__((ext_vector_type(8))) float v8f;

#define BB 4
#define NN 2048
#define DD 256
#define LL 256
#define K_BIAS 1.0f
#define HS_START 2.0f
#define INTERVAL 7.0f
#define ROWS (BB * NN)          // 8192
#define ADJ_ELEMS ((size_t)ROWS * NN)
#define INV_N (1.0f / (float)NN)

// ---------------------------------------------------------------------------
// Kernel 1: fold W_mu (D,L) @ W_kp (L,1) -> w_eff[D], and the scalar bias
//   c = b_mu @ W_kp + b_kp + K_BIAS.  One 256-thread block; trivial cost.
// ---------------------------------------------------------------------------
__global__ void fold_weights_kernel(const float* __restrict__ W_mu,
                                    const float* __restrict__ b_mu,
                                    const float* __restrict__ W_kp,
                                    const float* __restrict__ b_kp,
                                    float* __restrict__ ws) {
    int d = threadIdx.x;           // 0..255 == D
    float acc = 0.0f;
    const float* row = W_mu + (size_t)d * LL;
#pragma unroll 8
    for (int l = 0; l < LL; ++l) acc = fmaf(row[l], W_kp[l], acc);
    ws[d] = acc;                   // w_eff
    if (d == 0) {
        float c = 0.0f;
        for (int l = 0; l < LL; ++l) c = fmaf(b_mu[l], W_kp[l], c);
        ws[DD] = c + b_kp[0] + K_BIAS;
    }
}

// ---------------------------------------------------------------------------
// Kernel 2: k = x @ w_eff + c via V_WMMA_F32_16X16X4_F32.
// One wave (32 threads) per 16 rows; 64 WMMA steps accumulate K=256.
// A layout (ISA 7.12.2, 32-bit 16x4): lanes 0-15 hold M=lane, VGPR0=K0,
// VGPR1=K1; lanes 16-31 hold M=lane-16, VGPR0=K2, VGPR1=K3.
// B columns are all w_eff -> every D column equals x_row . w_eff.
// C layout: lane 0 holds M=0..7 in acc[0..7]; lane 16 holds M=8..15.
// ---------------------------------------------------------------------------
__global__ void gemv_k_wmma_kernel(const float* __restrict__ x,
                                   const float* __restrict__ ws,
                                   float* __restrict__ k_out) {
    const int lane = threadIdx.x;          // 32 threads per block = one wave
    const int half = lane >> 4;            // 0: K0/K1, 1: K2/K3
    const int r    = lane & 15;            // row within 16-row tile
    const int row  = blockIdx.x * 16 + r;  // global row in [0, 8192)

    const float* xrow = x + (size_t)row * DD;
    v8f acc = {};
#pragma unroll 4
    for (int k0 = 0; k0 < DD; k0 += 4) {
        const int kk = k0 + 2 * half;
        v2f a, b;
        a.x = xrow[kk];
        a.y = xrow[kk + 1];
        b.x = ws[kk];
        b.y = ws[kk + 1];
        acc = __builtin_amdgcn_wmma_f32_16x16x4_f32(
            /*neg_a=*/false, a, /*neg_b=*/false, b,
            /*c_mod=*/(short)0, acc, /*reuse_a=*/false, /*reuse_b=*/false);
    }

    const float cbias = ws[DD];
    const int base = blockIdx.x * 16;
    if (lane == 0) {
#pragma unroll
        for (int g = 0; g < 8; ++g) k_out[base + g] = acc[g] + cbias;
    } else if (lane == 16) {
#pragma unroll
        for (int g = 0; g < 8; ++g) k_out[base + 8 + g] = acc[g] + cbias;
    }
}

// ---------------------------------------------------------------------------
// Kernel 3 (bandwidth-bound, ~67 MB stores): for each row i,
//   adj[i, j] = (1/N) * sigmoid(HS_START - INTERVAL*j + (k[i]-1)*INTERVAL)
// One block per row, 256 threads, 8 elems/thread as two float4 stores.
// ---------------------------------------------------------------------------
__device__ __forceinline__ float fast_sigmoid(float z) {
    return __builtin_amdgcn_rcpf(1.0f + __expf(-z));
}

__global__ void adj_kernel(const float* __restrict__ k_vals,
                           float* __restrict__ adj) {
    const int row = blockIdx.x;                        // 0..8191
    const float base = HS_START + (k_vals[row] - 1.0f) * INTERVAL;
    float4* out = reinterpret_cast<float4*>(adj + (size_t)row * NN);
    const int t = threadIdx.x;                         // 0..255
#pragma unroll
    for (int v = 0; v < 2; ++v) {
        const int j0 = t * 8 + v * 4;
        float4 o;
        o.x = INV_N * fast_sigmoid(base - INTERVAL * (float)(j0 + 0));
        o.y = INV_N * fast_sigmoid(base - INTERVAL * (float)(j0 + 1));
        o.z = INV_N * fast_sigmoid(base - INTERVAL * (float)(j0 + 2));
        o.w = INV_N * fast_sigmoid(base - INTERVAL * (float)(j0 + 3));
        out[j0 >> 2] = o;
    }
}

// ---------------------------------------------------------------------------
// Launch: inputs in setup_inputs() order:
// 0:x 1:temp 2:W_in 3:b_in 4:W_d 5:b_d 6:W_mu 7:b_mu 8:W_kp 9:b_kp
// Output: adj (B*N*N floats) then k (B*N floats), concatenated flat.
// W_in/b_in/W_d/b_d/temp are dead code (softmax over singleton axis == 1).
// ---------------------------------------------------------------------------
extern "C" void kernel_launch(void* const* d_in, const int* in_sizes, int n_in,
                              void* d_out, int out_size, void* d_ws, size_t ws_size,
                              hipStream_t stream) {
    const float* x    = (const float*)d_in[0];
    const float* W_mu = (const float*)d_in[6];
    const float* b_mu = (const float*)d_in[7];
    const float* W_kp = (const float*)d_in[8];
    const float* b_kp = (const float*)d_in[9];

    float* out   = (float*)d_out;
    float* adj   = out;                 // (B, N, N)
    float* k_out = out + ADJ_ELEMS;     // (B, N, 1)
    float* ws    = (float*)d_ws;        // [0..255]=w_eff, [256]=c

    fold_weights_kernel<<<1, 256, 0, stream>>>(W_mu, b_mu, W_kp, b_kp, ws);
    gemv_k_wmma_kernel<<<ROWS / 16, 32, 0, stream>>>(x, ws, k_out);
    adj_kernel<<<ROWS, 256, 0, stream>>>(k_out, adj);
}